// OtherEmbedding_18468359373266
// MI455X (gfx1250) — compile-verified
//
#include <hip/hip_runtime.h>
#include <math.h>

#define HEADS 4
#define CH    32
#define OUTD  128
#define IND   64
#define NEG_SLOPE 0.2f

typedef float v2f __attribute__((ext_vector_type(2)));
typedef float v8f __attribute__((ext_vector_type(8)));

// float atomic-max via sign-split int trick (GLOBAL_ATOMIC_MAX_I32 / MIN_U32);
// correct for mixed signs with -inf initialization.
__device__ __forceinline__ void atomicMaxFloat(float* addr, float v) {
  if (v >= 0.0f) atomicMax((int*)addr, __float_as_int(v));
  else           atomicMin((unsigned int*)addr, __float_as_uint(v));
}

__global__ void fill_kernel(float* __restrict__ p, float v, int n) {
  int t = blockIdx.x * blockDim.x + threadIdx.x;
  if (t < n) p[t] = v;
}

// wt[d*4+h] = sum_c W[d*128 + h*32 + c] * att[h*32 + c]
__global__ void fold_att_kernel(const float* __restrict__ W,
                                const float* __restrict__ att,
                                float* __restrict__ wt) {
  int t = threadIdx.x;           // 256 threads, 1 block
  int d = t >> 2, h = t & 3;
  float acc = 0.f;
  #pragma unroll
  for (int c = 0; c < CH; ++c)
    acc += W[d * OUTD + h * CH + c] * att[h * CH + c];
  wt[t] = acc;
}

// a[n*4+h] = sum_d x[n*64+d] * wt[d*4+h]
__global__ void att_logit_kernel(const float* __restrict__ x,
                                 const float* __restrict__ wt,
                                 float* __restrict__ a, int n) {
  int t = blockIdx.x * blockDim.x + threadIdx.x;
  if (t >= n) return;
  const float* xr = x + (size_t)t * IND;
  float a0 = 0.f, a1 = 0.f, a2 = 0.f, a3 = 0.f;
  for (int d = 0; d < IND; ++d) {
    float xv = xr[d];
    a0 += xv * wt[d * 4 + 0];
    a1 += xv * wt[d * 4 + 1];
    a2 += xv * wt[d * 4 + 2];
    a3 += xv * wt[d * 4 + 3];
  }
  a[t * 4 + 0] = a0; a[t * 4 + 1] = a1; a[t * 4 + 2] = a2; a[t * 4 + 3] = a3;
}

// C[M,128] = A[M,64] @ W[64,128] in fp32 via V_WMMA_F32_16X16X4_F32.
// One wave per 16x16 tile; 8 waves/block -> 16 rows x 128 cols per block.
// Lane = half*16 + idx; K element index = vgpr + 2*half for both A and B;
// C vgpr r holds output row (r + 8*half), col idx. M is a multiple of 16
// for every call here, so EXEC is all-ones (WMMA requirement).
__global__ __launch_bounds__(256) void gemm_64x128_wmma(
    const float* __restrict__ A, const float* __restrict__ W,
    float* __restrict__ C, int M) {
  const int lane = threadIdx.x & 31;
  const int wave = threadIdx.x >> 5;      // column tile 0..7
  const int row0 = blockIdx.x * 16;
  const int col0 = wave * 16;
  const int half = lane >> 4;
  const int idx  = lane & 15;
  if (row0 >= M) return;                  // block-uniform; never splits a wave

  v8f acc = {0.f, 0.f, 0.f, 0.f, 0.f, 0.f, 0.f, 0.f};
  const float* arow = A + (size_t)(row0 + idx) * IND;
  #pragma unroll
  for (int kt = 0; kt < 16; ++kt) {
    const int kbase = kt * 4 + half * 2;
    v2f a, b;
    a.x = arow[kbase];
    a.y = arow[kbase + 1];
    b.x = W[(size_t)kbase * OUTD + col0 + idx];
    b.y = W[(size_t)(kbase + 1) * OUTD + col0 + idx];
    acc = __builtin_amdgcn_wmma_f32_16x16x4_f32(
        false, a, false, b, (short)0, acc, false, false);
  }
  #pragma unroll
  for (int r = 0; r < 8; ++r) {
    int m = r + half * 8;
    C[(size_t)(row0 + m) * OUTD + col0 + idx] = acc[r];
  }
}

__global__ void edge_logit_max_kernel(const int* __restrict__ src,
                                      const int* __restrict__ dst,
                                      const float* __restrict__ a_s,
                                      const float* __restrict__ a_d,
                                      float* __restrict__ e,
                                      float* __restrict__ m, int E) {
  int t = blockIdx.x * blockDim.x + threadIdx.x;
  if (t >= E * HEADS) return;
  int k = t >> 2, h = t & 3;
  int si = src[k], di = dst[k];
  float v = a_s[si * 4 + h] + a_d[di * 4 + h];
  v = (v > 0.f) ? v : NEG_SLOPE * v;       // leaky_relu
  e[t] = v;
  atomicMaxFloat(&m[di * 4 + h], v);
}

__global__ void segmax_fix_kernel(float* __restrict__ m, int n) {
  int t = blockIdx.x * blockDim.x + threadIdx.x;
  if (t >= n) return;
  float v = m[t];
  m[t] = isfinite(v) ? v : 0.f;            // empty segments -> 0
}

__global__ void edge_exp_sum_kernel(const int* __restrict__ dst,
                                    float* __restrict__ e,
                                    const float* __restrict__ m,
                                    float* __restrict__ s, int E) {
  int t = blockIdx.x * blockDim.x + threadIdx.x;
  if (t >= E * HEADS) return;
  int k = t >> 2, h = t & 3;
  int di = dst[k];
  float ex = expf(e[t] - m[di * 4 + h]);
  e[t] = ex;                               // overwrite logits with exp values
  atomicAdd(&s[di * 4 + h], ex);
}

// one wave per edge; each lane handles 4 consecutive channels (head = lane/8)
__global__ __launch_bounds__(256) void edge_msg_kernel(
    const int* __restrict__ src, const int* __restrict__ dst,
    const float* __restrict__ ex, const float* __restrict__ ssum,
    const float* __restrict__ hs, float* __restrict__ msg, int E) {
  int t = blockIdx.x * blockDim.x + threadIdx.x;
  int k = t >> 5;
  if (k >= E) return;
  int lane = t & 31;
  int h = lane >> 3;
  int si = src[k], di = dst[k];
  float alpha = ex[(size_t)k * 4 + h] / (ssum[di * 4 + h] + 1e-16f);
  const float4 hv = *((const float4*)(hs + (size_t)si * OUTD) + lane);
  float* mp = msg + (size_t)di * OUTD + lane * 4;
  atomicAdd(mp + 0, hv.x * alpha);
  atomicAdd(mp + 1, hv.y * alpha);
  atomicAdd(mp + 2, hv.z * alpha);
  atomicAdd(mp + 3, hv.w * alpha);
}

// out = LayerNorm(relu(msg + bias1 [+ bias2])) ; one wave per node
__global__ __launch_bounds__(256) void finalize_ln_kernel(
    const float* __restrict__ msg, const float* __restrict__ b1,
    const float* __restrict__ b2, const float* __restrict__ g,
    const float* __restrict__ beta, float* __restrict__ out, int N) {
  int t = blockIdx.x * blockDim.x + threadIdx.x;
  int node = t >> 5;
  if (node >= N) return;
  int lane = t & 31;
  float v[4];
  float sum = 0.f, sumsq = 0.f;
  #pragma unroll
  for (int i = 0; i < 4; ++i) {
    int c = lane * 4 + i;
    float x = msg[(size_t)node * OUTD + c] + b1[c] + (b2 ? b2[c] : 0.f);
    x = (x > 0.f) ? x : 0.f;               // relu
    v[i] = x; sum += x; sumsq += x * x;
  }
  #pragma unroll
  for (int o = 16; o > 0; o >>= 1) {
    sum   += __shfl_xor(sum, o, 32);
    sumsq += __shfl_xor(sumsq, o, 32);
  }
  float mu  = sum * (1.f / OUTD);
  float var = sumsq * (1.f / OUTD) - mu * mu;
  float inv = rsqrtf(var + 1e-5f);
  #pragma unroll
  for (int i = 0; i < 4; ++i) {
    int c = lane * 4 + i;
    out[(size_t)node * OUTD + c] = (v[i] - mu) * inv * g[c] + beta[c];
  }
}

static inline int cdiv(int a, int b) { return (a + b - 1) / b; }

extern "C" void kernel_launch(void* const* d_in, const int* in_sizes, int n_in,
                              void* d_out, int out_size, void* d_ws, size_t ws_size,
                              hipStream_t stream) {
  const float* x_job     = (const float*)d_in[0];
  const float* x_station = (const float*)d_in[1];
  const float* x_machine = (const float*)d_in[2];
  const float* x_robot   = (const float*)d_in[3];
  const int* edges[5] = { (const int*)d_in[4], (const int*)d_in[5],
                          (const int*)d_in[6], (const int*)d_in[7],
                          (const int*)d_in[8] };

  const int Nj = in_sizes[0] / IND;
  const int Ns = in_sizes[1] / IND;
  const int Nm = in_sizes[2] / IND;
  const int Nr = in_sizes[3] / IND;
  const int E  = in_sizes[4] / 2;
  (void)n_in; (void)out_size; (void)ws_size;

  // params flattened in insertion order after the 9 tensor inputs:
  // 5 GAT dicts (W_src, W_dst, att_src, att_dst, bias) for cbl,li,nd,eb,hb;
  // then res_station/machine/robot; then ln g/b for station, machine, robot.
  const float* P[34];
  for (int i = 0; i < 34; ++i) P[i] = (const float*)d_in[9 + i];
  const float* res_W[3] = { P[25], P[26], P[27] };
  const float* ln_g[3]  = { P[28], P[30], P[32] };
  const float* ln_b[3]  = { P[29], P[31], P[33] };

  int Nmax = Ns > Nm ? Ns : Nm; if (Nr > Nmax) Nmax = Nr;

  // workspace layout (floats), ~142 MB
  float* w = (float*)d_ws;
  size_t off = 0;
  float* hs   = w + off; off += (size_t)Nj * OUTD;
  float* a_s  = w + off; off += (size_t)Nj * HEADS;
  float* a_d  = w + off; off += (size_t)Nmax * HEADS;
  float* ebuf = w + off; off += (size_t)E * HEADS;
  float* mbuf = w + off; off += (size_t)Nmax * HEADS;
  float* sbuf = w + off; off += (size_t)Nmax * HEADS;
  float* msg  = w + off; off += (size_t)Nmax * OUTD;
  float* wt_s = w + off; off += 256;
  float* wt_d = w + off; off += 256;

  const float NEG_INF = -__builtin_inff();

  const float* x_dst_of[3] = { x_station, x_machine, x_robot };
  const int    N_of[3]     = { Ns, Nm, Nr };
  const int    rel_of[3][2] = { {0, 1}, {2, 3}, {4, -1} };  // cbl+li | nd+eb | hb

  float* out_ptr = (float*)d_out;
  for (int ty = 0; ty < 3; ++ty) {
    const int Ndst = N_of[ty];
    const float* x_dst = x_dst_of[ty];

    // seed msg with the residual projection: msg = x_dst @ res_W  (WMMA)
    gemm_64x128_wmma<<<Ndst / 16, 256, 0, stream>>>(x_dst, res_W[ty], msg, Ndst);

    for (int ri = 0; ri < 2; ++ri) {
      int r = rel_of[ty][ri];
      if (r < 0) break;
      const float* Wsrc = P[r * 5 + 0];
      const float* Wdst = P[r * 5 + 1];
      const float* atts = P[r * 5 + 2];
      const float* attd = P[r * 5 + 3];
      const int* src = edges[r];
      const int* dst = edges[r] + E;

      fold_att_kernel<<<1, 256, 0, stream>>>(Wsrc, atts, wt_s);
      fold_att_kernel<<<1, 256, 0, stream>>>(Wdst, attd, wt_d);
      att_logit_kernel<<<cdiv(Nj, 256), 256, 0, stream>>>(x_job, wt_s, a_s, Nj);
      att_logit_kernel<<<cdiv(Ndst, 256), 256, 0, stream>>>(x_dst, wt_d, a_d, Ndst);

      // hs = x_job @ W_src  (WMMA)
      gemm_64x128_wmma<<<Nj / 16, 256, 0, stream>>>(x_job, Wsrc, hs, Nj);

      // per-destination softmax over edges
      fill_kernel<<<cdiv(Ndst * HEADS, 256), 256, 0, stream>>>(mbuf, NEG_INF, Ndst * HEADS);
      fill_kernel<<<cdiv(Ndst * HEADS, 256), 256, 0, stream>>>(sbuf, 0.f, Ndst * HEADS);
      edge_logit_max_kernel<<<cdiv(E * HEADS, 256), 256, 0, stream>>>(
          src, dst, a_s, a_d, ebuf, mbuf, E);
      segmax_fix_kernel<<<cdiv(Ndst * HEADS, 256), 256, 0, stream>>>(mbuf, Ndst * HEADS);
      edge_exp_sum_kernel<<<cdiv(E * HEADS, 256), 256, 0, stream>>>(dst, ebuf, mbuf, sbuf, E);
      edge_msg_kernel<<<cdiv(E * 32, 256), 256, 0, stream>>>(src, dst, ebuf, sbuf, hs, msg, E);
    }

    const float* b1 = P[rel_of[ty][0] * 5 + 4];
    const float* b2 = (rel_of[ty][1] >= 0) ? P[rel_of[ty][1] * 5 + 4] : nullptr;
    finalize_ln_kernel<<<cdiv(Ndst * 32, 256), 256, 0, stream>>>(
        msg, b1, b2, ln_g[ty], ln_b[ty], out_ptr, Ndst);
    out_ptr += (size_t)Ndst * OUTD;
  }
}